// Conv1dFFTPrimative_39256001085523
// MI455X (gfx1250) — compile-verified
//
#include <hip/hip_runtime.h>
#include <math.h>
#include <stdint.h>

// ============================================================================
// Conv1dFFT on MI455X (gfx1250):
//   per batch row (B=256, T=65536):
//     X = FFT_65536(x); Xf = X * filt; fold by 8; y = IFFT_8192; out = (re,im)
//
// Four-step decomposition, all 16-point DFT butterflies as fp32 WMMA
// (v_wmma_f32_16x16x4_f32, K accumulated over 4 slices):
//   S1: column FFT_256 over a (n = 256a + b'), twiddle w_65536^{b'c} -> ws0
//       column tile staged to LDS via TDM tensor_load_to_lds (2D descriptor)
//   S2: row FFT_256 over b'  -> X[256d + c]  (row staged to LDS via
//       global_load_async_to_lds_b128), * filt, fold d = e+32j -> Y (1/8 sum)
//   S3: 32-pt IDFT over e (VALU), twiddle wbar_8192^{c n1} -> ws1[c*32+n1]
//   S4: 256-pt IFFT over c (per-wave TDM gather of the strided column)
//       -> y[n1 + 32 n2], scale 1/8192 -> out
//
// Each 256-FFT runs entirely inside ONE wave32: two complex 16x16x16 matmuls
// (16 WMMA each) with a per-wave LDS transpose between stages.
// Workspace: ws0 = 256*65536 float2 (128 MiB), ws1 = 256*8192 float2 (16 MiB).
// ============================================================================

typedef float v2f __attribute__((ext_vector_type(2)));
typedef float v8f __attribute__((ext_vector_type(8)));
typedef unsigned int v4u __attribute__((ext_vector_type(4)));
typedef unsigned int v8u __attribute__((ext_vector_type(8)));

#define TWO_PI 6.28318530717958647692f

__device__ __forceinline__ v8f wmma_f32(v2f a, v2f b, v8f c) {
  // D = A(16x4 f32) * B(4x16 f32) + C(16x16 f32)
  return __builtin_amdgcn_wmma_f32_16x16x4_f32(
      /*neg_a=*/false, a, /*neg_b=*/false, b,
      /*c_mod=*/(short)0, c, /*reuse_a=*/false, /*reuse_b=*/false);
}

// ---------------------------------------------------------------------------
// TDM: issue a 2D tile load (Global -> LDS) via a D# descriptor in SGPRs.
// Group0 (ISA 8.3): count=1, lds_addr, global_addr[56:0], type=2.
// Group1 (ISA 8.4): workgroup_mask=0 (no cluster), data_size=2 (4B),
//   tensor_dim0/1 == tile_dim0/1 (no OOB), tensor_dim0_stride in elements.
// ---------------------------------------------------------------------------
__device__ __forceinline__ void tdm_load_2d(uint32_t lds_addr, uint64_t gaddr,
                                            uint32_t tile_d0, uint32_t tile_d1,
                                            uint64_t stride0_elems) {
  v4u g0;
  g0[0] = 1u;                                   // count=1 (valid), no gather
  g0[1] = lds_addr;                             // LDS byte address
  g0[2] = (uint32_t)gaddr;                      // global_addr[31:0]
  g0[3] = (uint32_t)((gaddr >> 32) & 0x1FFFFFFu) | (2u << 30); // [56:32]|type=2
  v8u g1;
  g1[0] = (2u << 16);                           // data_size = 4 bytes
  g1[1] = (tile_d0 & 0xFFFFu) << 16;            // tensor_dim0[15:0] (=tile_d0)
  g1[2] = ((tile_d0 >> 16) & 0xFFFFu) | ((tile_d1 & 0xFFFFu) << 16);
  g1[3] = ((tile_d1 >> 16) & 0xFFFFu) | ((tile_d0 & 0xFFFFu) << 16); // tile_dim0
  g1[4] = (tile_d1 & 0xFFFFu);                  // tile_dim1, tile_dim2=0
  g1[5] = (uint32_t)stride0_elems;              // tensor_dim0_stride[31:0]
  g1[6] = (uint32_t)((stride0_elems >> 32) & 0xFFFFu); // stride0[47:32]
  g1[7] = 0u;                                   // tensor_dim1_stride (unused 2D)
  asm volatile("tensor_load_to_lds %0, %1" ::"s"(g0), "s"(g1) : "memory");
}

// A-layout slices of the 16x16 DFT matrix D[m][k] = exp(i * sgn * 2pi * m*k/16)
__device__ __forceinline__ void build_d16(float sgn, v2f ar[4], v2f ai[4]) {
  const int lane = threadIdx.x & 31;
  const int m = lane & 15;
  const int off = (lane >> 4) << 1; // 0 or 2
#pragma unroll
  for (int k = 0; k < 4; ++k) {
    const int k0 = 4 * k + off;
    float s0, c0, s1, c1;
    __sincosf(sgn * (TWO_PI / 16.0f) * (float)(m * k0), &s0, &c0);
    __sincosf(sgn * (TWO_PI / 16.0f) * (float)(m * (k0 + 1)), &s1, &c1);
    ar[k][0] = c0; ar[k][1] = c1;
    ai[k][0] = s0; ai[k][1] = s1;
  }
}

// Complex 16x16x16 matmul via 16 x v_wmma_f32_16x16x4_f32.
__device__ __forceinline__ void cmma(const v2f ar[4], const v2f ai[4],
                                     const v2f br[4], const v2f bi[4],
                                     v8f& outr, v8f& outi) {
  v8f rr = {}, ii = {}, ri = {}, ir = {};
#pragma unroll
  for (int k = 0; k < 4; ++k) {
    rr = wmma_f32(ar[k], br[k], rr);
    ii = wmma_f32(ai[k], bi[k], ii);
    ri = wmma_f32(ar[k], bi[k], ri);
    ir = wmma_f32(ai[k], br[k], ir);
  }
  outr = rr - ii;
  outi = ri + ir;
}

// Wave-level 256-point (I)FFT of v[a], a = a2 + 16*a1, sgn = -1 fwd / +1 inv.
// Input : B-layout complex tiles of V[a1][a2]  (br/bi, 4 K-slices)
// Output: C-layout F^T[c2][c1] where F[c1 + 16 c2] is the transform.
__device__ __forceinline__ void fft256_wave(float sgn, const v2f br[4],
                                            const v2f bi[4], float (*tile)[2],
                                            v8f& fr, v8f& fi) {
  v2f ar[4], ai[4];
  build_d16(sgn, ar, ai);

  v8f t1r, t1i;
  cmma(ar, ai, br, bi, t1r, t1i); // T1 = D * V   (sum over a1)

  const int lane = threadIdx.x & 31;
  const int col = lane & 15;            // a2
  const int rb = (lane >> 4) << 3;      // row base 0/8
#pragma unroll
  for (int r = 0; r < 8; ++r) {
    const int row = r + rb;             // c1
    float s, c;
    __sincosf(sgn * (TWO_PI / 256.0f) * (float)(row * col), &s, &c);
    const float xr = t1r[r], xi = t1i[r];
    tile[row * 16 + col][0] = xr * c - xi * s; // T2 = T1 .* w_256^{c1*a2}
    tile[row * 16 + col][1] = xr * s + xi * c;
  }
  asm volatile("s_wait_dscnt 0" ::: "memory"); // LDS RAW fence (wave-private)

  // Read T2^T back as B-layout; F^T = D * T2^T (D symmetric).
  v2f br2[4], bi2[4];
  const int off = (lane >> 4) << 1;
#pragma unroll
  for (int k = 0; k < 4; ++k) {
    const int r0 = 4 * k + off;
    br2[k][0] = tile[col * 16 + r0][0];
    br2[k][1] = tile[col * 16 + r0 + 1][0];
    bi2[k][0] = tile[col * 16 + r0][1];
    bi2[k][1] = tile[col * 16 + r0 + 1][1];
  }
  cmma(ar, ai, br2, bi2, fr, fi);
}

// ---------------------------------------------------------------------------
// S1: column FFT_256 + outer twiddle.  grid(32,256), block 256 (8 waves).
// TDM stages the 256x8 column tile (1KB-stride rows) into LDS in one DMA.
// ---------------------------------------------------------------------------
__global__ __launch_bounds__(256) void k_fft_cols(const float* __restrict__ x,
                                                  float2* __restrict__ ws0) {
  __shared__ float ldsX[256][8];     // [a][col-in-tile]
  __shared__ float tile[8][256][2];
  const int w = threadIdx.x >> 5;
  const int lane = threadIdx.x & 31;
  const int b = blockIdx.y;
  const int colbase = blockIdx.x * 8;
  const int colg = colbase + w; // [0,256)
  const float* xb = x + (size_t)b * 65536;

  if (w == 0) { // one TDM descriptor per block: tile_dim = 8 x 256, stride 256
    tdm_load_2d((uint32_t)(uintptr_t)&ldsX[0][0],
                (uint64_t)(uintptr_t)(xb + colbase),
                /*tile_d0=*/8u, /*tile_d1=*/256u, /*stride0=*/256u);
    __builtin_amdgcn_s_wait_tensorcnt(0);
  }
  __syncthreads();

  const int col16 = lane & 15;
  const int off = (lane >> 4) << 1;
  v2f br[4], bi[4];
#pragma unroll
  for (int k = 0; k < 4; ++k) {
    const int a1 = 4 * k + off;
    const int e0 = col16 + 16 * a1;       // v[a] = x[a*256 + colg]
    const int e1 = col16 + 16 * (a1 + 1);
    br[k][0] = ldsX[e0][w];
    br[k][1] = ldsX[e1][w];
    bi[k][0] = 0.0f; bi[k][1] = 0.0f;     // input is real
  }

  v8f fr, fi;
  fft256_wave(-1.0f, br, bi, tile[w], fr, fi);

  float2* wsb = ws0 + (size_t)b * 65536;
  const int rb = (lane >> 4) << 3;
#pragma unroll
  for (int r = 0; r < 8; ++r) {
    const int c = col16 + 16 * (r + rb);
    float s, ct;
    __sincosf(-(TWO_PI / 65536.0f) * (float)(colg * c), &s, &ct);
    const float xr = fr[r], xi = fi[r];
    float2 v; v.x = xr * ct - xi * s; v.y = xr * s + xi * ct;
    wsb[(size_t)c * 256 + colg] = v; // B2[c][colg]
  }
}

// ---------------------------------------------------------------------------
// S2+fold+S3: row FFT_256, *filt, fold-by-8, 32-pt IDFT, twiddle.
// Row (2KB contiguous) staged per-wave with global_load_async_to_lds_b128.
// ---------------------------------------------------------------------------
__global__ __launch_bounds__(256) void k_fft_rows_fold(
    const float2* __restrict__ ws0, const float* __restrict__ filt,
    float2* __restrict__ ws1) {
  __shared__ float ldsRow[8][256][2]; // per-wave staged row (2KB each)
  __shared__ float tile[8][256][2];
  __shared__ float yfold[8][32][2];
  const int w = threadIdx.x >> 5;
  const int lane = threadIdx.x & 31;
  const int b = blockIdx.y;
  const int c = blockIdx.x * 8 + w;
  const float2* row = ws0 + (size_t)b * 65536 + (size_t)c * 256;
  __builtin_prefetch(filt + c, 0, 0); // global_prefetch_b8

  { // async copy: 4 x (32 lanes x 16B) = 2KB, wave-private -> no barrier
    const uint32_t ldsb = (uint32_t)(uintptr_t)&ldsRow[w][0][0];
#pragma unroll
    for (int i = 0; i < 4; ++i) {
      const uint32_t goff = (uint32_t)(lane * 16 + i * 512);
      const uint32_t loff = ldsb + (uint32_t)(lane * 16 + i * 512);
      asm volatile("global_load_async_to_lds_b128 %0, %1, %2"
                   ::"v"(loff), "v"(goff), "s"(row)
                   : "memory");
    }
    asm volatile("s_wait_asynccnt 0" ::: "memory");
  }

  const int col16 = lane & 15;
  const int off = (lane >> 4) << 1;
  v2f br[4], bi[4];
#pragma unroll
  for (int k = 0; k < 4; ++k) {
    const int bidx = 4 * k + off;
    const int e0 = col16 + 16 * bidx;
    const int e1 = col16 + 16 * (bidx + 1);
    br[k][0] = ldsRow[w][e0][0]; br[k][1] = ldsRow[w][e1][0];
    bi[k][0] = ldsRow[w][e0][1]; bi[k][1] = ldsRow[w][e1][1];
  }

  v8f fr, fi;
  fft256_wave(-1.0f, br, bi, tile[w], fr, fi); // X[256 d + c], d = col+16*row

  const int rb = (lane >> 4) << 3;
#pragma unroll
  for (int r = 0; r < 8; ++r) {
    const int d = col16 + 16 * (r + rb);
    const float g = filt[(size_t)d * 256 + c];
    tile[w][d][0] = fr[r] * g; // Xf linear by d in the wave tile
    tile[w][d][1] = fi[r] * g;
  }
  asm volatile("s_wait_dscnt 0" ::: "memory");

  // fold: Y[e] = (1/8) sum_q Xf[e + 32 q], lane = e in [0,32)
  {
    float sr = 0.0f, si = 0.0f;
#pragma unroll
    for (int q = 0; q < 8; ++q) {
      sr += tile[w][lane + 32 * q][0];
      si += tile[w][lane + 32 * q][1];
    }
    yfold[w][lane][0] = sr * 0.125f;
    yfold[w][lane][1] = si * 0.125f;
  }
  asm volatile("s_wait_dscnt 0" ::: "memory");

  // 32-pt IDFT over e (lane = n1), then twiddle wbar_8192^{c*n1}
  float pr = 0.0f, pi = 0.0f;
#pragma unroll
  for (int e = 0; e < 32; ++e) {
    float s, ch;
    __sincosf((TWO_PI / 32.0f) * (float)(e * lane), &s, &ch);
    const float yr = yfold[w][e][0], yi = yfold[w][e][1];
    pr += yr * ch - yi * s;
    pi += yr * s + yi * ch;
  }
  float s, ch;
  __sincosf((TWO_PI / 8192.0f) * (float)(c * lane), &s, &ch);
  float2 v; v.x = pr * ch - pi * s; v.y = pr * s + pi * ch;
  ws1[(size_t)b * 8192 + c * 32 + lane] = v; // P[c][n1]
}

// ---------------------------------------------------------------------------
// S4: 256-pt IFFT over c per n1; y[n1 + 32 n2] /= 8192. grid(4,256), block 256.
// Strided column gather (8B every 256B) done with a per-wave TDM descriptor.
// ---------------------------------------------------------------------------
__global__ __launch_bounds__(256) void k_ifft_final(
    const float2* __restrict__ ws1, float* __restrict__ out) {
  __shared__ float ldsU[8][256][2]; // per-wave gathered column (2KB each)
  __shared__ float tile[8][256][2];
  const int w = threadIdx.x >> 5;
  const int lane = threadIdx.x & 31;
  const int b = blockIdx.y;
  const int n1 = blockIdx.x * 8 + w; // [0,32)
  const float2* pb = ws1 + (size_t)b * 8192;

  // per-wave TDM: tile 2 x 256 f32 (one complex per row), row stride 64 f32
  tdm_load_2d((uint32_t)(uintptr_t)&ldsU[w][0][0],
              (uint64_t)(uintptr_t)(pb + n1),
              /*tile_d0=*/2u, /*tile_d1=*/256u, /*stride0=*/64u);
  __builtin_amdgcn_s_wait_tensorcnt(0);

  const int col16 = lane & 15;
  const int off = (lane >> 4) << 1;
  v2f br[4], bi[4];
#pragma unroll
  for (int k = 0; k < 4; ++k) {
    const int cidx = 4 * k + off;
    const int c0 = col16 + 16 * cidx;
    const int c1 = col16 + 16 * (cidx + 1);
    br[k][0] = ldsU[w][c0][0]; br[k][1] = ldsU[w][c1][0];
    bi[k][0] = ldsU[w][c0][1]; bi[k][1] = ldsU[w][c1][1];
  }

  v8f fr, fi;
  fft256_wave(+1.0f, br, bi, tile[w], fr, fi); // n2 = col + 16*row

  float* ob = out + (size_t)b * 16384; // [8192][2]
  const int rb = (lane >> 4) << 3;
  const float sc = 1.0f / 8192.0f;
#pragma unroll
  for (int r = 0; r < 8; ++r) {
    const int n2 = col16 + 16 * (r + rb);
    const int n = n1 + 32 * n2;
    ob[2 * n + 0] = fr[r] * sc;
    ob[2 * n + 1] = fi[r] * sc;
  }
}

// ---------------------------------------------------------------------------
extern "C" void kernel_launch(void* const* d_in, const int* in_sizes, int n_in,
                              void* d_out, int out_size, void* d_ws,
                              size_t ws_size, hipStream_t stream) {
  const float* x = (const float*)d_in[0];    // [256,1,65536] f32
  const float* filt = (const float*)d_in[1]; // [65536] f32
  (void)in_sizes; (void)n_in; (void)out_size; (void)ws_size;
  // sampling_factor (d_in[2]) is fixed at 3 -> k = 8 baked into the fold.

  float* out = (float*)d_out;                // [256,1,8192,2] f32
  float2* ws0 = (float2*)d_ws;               // 256*65536 complex (128 MiB)
  float2* ws1 = ws0 + (size_t)256 * 65536;   // 256*8192 complex  (16 MiB)

  const dim3 blk(256, 1, 1);
  k_fft_cols<<<dim3(32, 256), blk, 0, stream>>>(x, ws0);
  k_fft_rows_fold<<<dim3(32, 256), blk, 0, stream>>>(ws0, filt, ws1);
  k_ifft_final<<<dim3(4, 256), blk, 0, stream>>>(ws1, out);
}